// ProbabilityField_84439057039541
// MI455X (gfx1250) — compile-verified
//
#include <hip/hip_runtime.h>

typedef float v2f __attribute__((ext_vector_type(2)));
typedef float v8f __attribute__((ext_vector_type(8)));

#define CHUNK 2048           // leaves per block in scan kernels (8 waves x 256)
#define GUIDE_MAX (4 << 20)  // up to 4M guide buckets (16 MB, L2-resident)

__device__ __forceinline__ float leafWeight(float w, int lvl) {
    // weights = leaf_weights * 2^{-level/2}
    return w * __builtin_exp2f(-0.5f * (float)lvl);
}

// ---------------------------------------------------------------------------
// Kernel 1: per-block reduction of transformed weights -> blockSums
// ---------------------------------------------------------------------------
__global__ void k_block_reduce(const float* __restrict__ lw, const int* __restrict__ lv,
                               float* __restrict__ bSums, int N) {
    __shared__ float red[256];
    const int t = threadIdx.x;
    const long long base = (long long)blockIdx.x * CHUNK;
    float s = 0.f;
#pragma unroll
    for (int j = 0; j < 8; ++j) {
        long long g = base + j * 256 + t;
        if (g < N) s += leafWeight(lw[g], lv[g]);
    }
    red[t] = s;
    __syncthreads();
    for (int off = 128; off > 0; off >>= 1) {
        if (t < off) red[t] += red[t + off];
        __syncthreads();
    }
    if (t == 0) bSums[blockIdx.x] = red[0];
}

// ---------------------------------------------------------------------------
// Kernel 2: single-block Hillis-Steele scan of blockSums -> exclusive offsets
//           (nB <= 1024 for N = 2M with CHUNK = 2048)
// ---------------------------------------------------------------------------
__global__ void k_scan_sums(const float* __restrict__ bSums, float* __restrict__ bOff,
                            float* __restrict__ totalP, int nB) {
    __shared__ float lds[1024];
    const int t = threadIdx.x;
    const float x = (t < nB) ? bSums[t] : 0.f;
    lds[t] = x;
    __syncthreads();
    for (int off = 1; off < 1024; off <<= 1) {
        float y = (t >= off) ? lds[t - off] : 0.f;
        __syncthreads();
        lds[t] += y;
        __syncthreads();
    }
    if (t < nB) bOff[t] = lds[t] - x;   // exclusive prefix
    if (t == nB - 1) *totalP = lds[t];  // grand total
}

// ---------------------------------------------------------------------------
// Kernel 3: per-block inclusive scan -> cdf, using V_WMMA_F32_16X16X4_F32.
// Each wave scans 256 values: D = L * X where L is 16x16 lower-triangular
// ones, realized as 4 chained K=4 WMMAs (full f32, exact).
//   A 16x4 layout: lanes 0-15 hold M=lane K={0,1}; lanes 16-31 M=lane-16 K={2,3}
//   B 4x16 layout: lanes 0-15 rows {k0,k0+1}=..{4m,4m+1}; lanes 16-31 {4m+2,4m+3}
//   C/D:           lane column n=lane&15, VGPR v holds row v+8*(lane>=16)
// ---------------------------------------------------------------------------
__global__ void k_block_scan(const float* __restrict__ lw, const int* __restrict__ lv,
                             const float* __restrict__ bOff,
                             float* __restrict__ cdf, int N) {
    __shared__ __align__(16) float tile[CHUNK];
    __shared__ float chunkTot[8][16];
    __shared__ float waveTot[8];

    const int t = threadIdx.x;
    const long long base = (long long)blockIdx.x * CHUNK;

    // Stage transformed weights into LDS (zero-padded tail).
#pragma unroll
    for (int j = 0; j < 8; ++j) {
        long long g = base + j * 256 + t;
        tile[j * 256 + t] = (g < N) ? leafWeight(lw[g], lv[g]) : 0.f;
    }
    __syncthreads();

    const int w  = t >> 5;        // wave id (wave32)
    const int ln = t & 31;        // lane id
    const int n  = ln & 15;       // chunk column (N dim) / A row (M dim)
    const int hi = ln >> 4;       // lane-half selector
    const float* gsrc = &tile[w * 256 + n * 16];

    v8f acc = {};
#pragma unroll
    for (int m = 0; m < 4; ++m) {
        const int k0 = 4 * m + 2 * hi;
        v2f a, b;
        a.x = (k0     <= n) ? 1.f : 0.f;   // L[n][k0]
        a.y = (k0 + 1 <= n) ? 1.f : 0.f;   // L[n][k0+1]
        b = *(const v2f*)&gsrc[k0];        // X[k0..k0+1][n] (8B-aligned LDS)
        acc = __builtin_amdgcn_wmma_f32_16x16x4_f32(
            /*neg_a=*/false, a, /*neg_b=*/false, b,
            /*c_mod=*/(short)0, acc, /*reuse_a=*/false, /*reuse_b=*/false);
    }
    // acc[v] = inclusive scan of 16-chunk n at position v + 8*hi.

    if (hi) chunkTot[w][n] = acc[7];       // row 15 = chunk total
    __syncthreads();
    if (ln == 0) {
        float s = 0.f;
#pragma unroll
        for (int j = 0; j < 16; ++j) s += chunkTot[w][j];
        waveTot[w] = s;
    }
    __syncthreads();

    float off = bOff[blockIdx.x];
    for (int j = 0; j < w; ++j) off += waveTot[j];
    for (int j = 0; j < n; ++j) off += chunkTot[w][j];

    const long long ebase = base + w * 256 + n * 16 + 8 * hi;
#pragma unroll
    for (int v = 0; v < 8; ++v) {
        long long g = ebase + v;
        if (g < N) cdf[g] = acc[v] + off;
    }
}

// ---------------------------------------------------------------------------
// Kernel 4: invert the CDF into a guide table. Leaf i owns buckets b with
// b*total/K in [cdf[i-1], cdf[i]); consecutive leaves evaluate the identical
// ceil expression, so the partition is gap-free even under f32 rounding.
// ---------------------------------------------------------------------------
__global__ void k_build_guide(const float* __restrict__ cdf, const float* __restrict__ totalP,
                              int* __restrict__ guide, int N, int K) {
    long long i = (long long)blockIdx.x * blockDim.x + threadIdx.x;
    if (i >= N) return;
    const float scale = (float)K / *totalP;
    const float cPrev = (i == 0) ? 0.f : cdf[i - 1];
    int bLo = (int)ceilf(cPrev * scale);
    int bHi = (i == N - 1) ? K : (int)ceilf(cdf[i] * scale);
    if (bLo < 0) bLo = 0;
    if (bHi > K) bHi = K;
    const int ii = (int)i;
    for (int b = bLo; b < bHi; ++b) guide[b] = ii;
}

// ---------------------------------------------------------------------------
// Kernel 5: sampling. Guide start + bidirectional linear fixup (exactly
// reproduces searchsorted-right + clip, independent of guide quality).
// ---------------------------------------------------------------------------
__global__ void k_sample(const float* __restrict__ cdf, const float* __restrict__ totalP,
                         const int* __restrict__ guide, int K,
                         const float* __restrict__ lc, const int* __restrict__ lv,
                         const float* __restrict__ ics,
                         const float* __restrict__ us, const float* __restrict__ uj,
                         float* __restrict__ outS, int* __restrict__ outI,
                         int N, long long S) {
    long long s = (long long)blockIdx.x * blockDim.x + threadIdx.x;
    if (s >= S) return;
    const float total = *totalP;
    const float u = us[s];
    const float v = u * total;

    int i;
    if (K > 0) {
        int b = (int)(u * (float)K);
        b = min(max(b, 0), K - 1);
        i = guide[b];
        i = min(max(i, 0), N - 1);
        while (i < N - 1 && cdf[i] <= v) ++i;   // forward to first cdf > v
        while (i > 0 && cdf[i - 1] > v) --i;    // backward fixup (rare)
    } else {
        int lo = 0, hi = N;                      // fallback: plain binary search
        while (lo < hi) {
            int mid = (lo + hi) >> 1;
            if (cdf[mid] <= v) lo = mid + 1; else hi = mid;
        }
        i = min(lo, N - 1);
    }

    const int lvl = lv[i];
    const float csx = ldexpf(ics[0], -lvl);
    const float csy = ldexpf(ics[1], -lvl);
    const float csz = ldexpf(ics[2], -lvl);

    const long long s3 = 3 * s;
    const long long i3 = 3LL * i;
    outS[s3 + 0] = lc[i3 + 0] + (uj[s3 + 0] - 0.5f) * csx;
    outS[s3 + 1] = lc[i3 + 1] + (uj[s3 + 1] - 0.5f) * csy;
    outS[s3 + 2] = lc[i3 + 2] + (uj[s3 + 2] - 0.5f) * csz;
    outI[s] = i;  // reference idx dtype is int32; stored as raw int32 bits
}

// ---------------------------------------------------------------------------
extern "C" void kernel_launch(void* const* d_in, const int* in_sizes, int n_in,
                              void* d_out, int out_size, void* d_ws, size_t ws_size,
                              hipStream_t stream) {
    const float* lc  = (const float*)d_in[0];  // leaf_centers [N,3]
    const int*   lv  = (const int*)  d_in[1];  // leaf_levels  [N]
    const float* lw  = (const float*)d_in[2];  // leaf_weights [N]
    const float* ics = (const float*)d_in[3];  // initial_cell_size [3]
    const float* us  = (const float*)d_in[4];  // u_sample [S]
    const float* uj  = (const float*)d_in[5];  // u_jitter [S,3]

    const int       N = in_sizes[1];
    const long long S = in_sizes[4];
    const int      nB = (N + CHUNK - 1) / CHUNK;   // <= 1024 for N = 2M

    // Workspace layout: cdf[N] | blockSums[nB] | blockOff[nB] | total | guide[K]
    float* cdf    = (float*)d_ws;
    float* bSums  = cdf + N;
    float* bOff   = bSums + nB;
    float* totalP = bOff + nB;
    int*   guide  = (int*)(totalP + 1);

    const size_t used = (size_t)(N + 2 * nB + 1) * sizeof(float);
    long long avail = (ws_size > used) ? (long long)((ws_size - used) / sizeof(int)) : 0;
    int K = (int)((avail < (long long)GUIDE_MAX) ? avail : (long long)GUIDE_MAX);
    if (K < 1024) K = 0;  // too little scratch: sampler falls back to binary search

    k_block_reduce<<<nB, 256, 0, stream>>>(lw, lv, bSums, N);
    k_scan_sums  <<<1, 1024, 0, stream>>>(bSums, bOff, totalP, nB);
    k_block_scan <<<nB, 256, 0, stream>>>(lw, lv, bOff, cdf, N);
    if (K > 0) {
        k_build_guide<<<(N + 255) / 256, 256, 0, stream>>>(cdf, totalP, guide, N, K);
    }

    float* outS = (float*)d_out;
    int*   outI = (int*)d_out + (size_t)3 * S;  // idx stored after samples
    const int blocksS = (int)((S + 255) / 256);
    k_sample<<<blocksS, 256, 0, stream>>>(cdf, totalP, guide, K, lc, lv, ics,
                                          us, uj, outS, outI, N, S);
}